// OnlyMambaModule_33062658244722
// MI455X (gfx1250) — compile-verified
//
#include <hip/hip_runtime.h>

// ---------------------------------------------------------------------------
// Mamba block for MI455X (gfx1250, wave32, WMMA + async global->LDS).
// GEMMs: V_WMMA_F32_16X16X4_F32 (fp32, matches reference dtype), 16x64 block
// tile (4 waves x 16x16), K-tile 32, double-buffered LDS tiles filled with
// GLOBAL_LOAD_ASYNC_TO_LDS_B128 (ASYNCcnt) so tile k+1 streams in while tile
// k is consumed by WMMAs. Scan: 1 thread/channel, 16 fp32 states in regs,
// B/C staged to LDS in double-buffered 64-step async chunks.
// ---------------------------------------------------------------------------

#define D_MODEL 512
#define D_INNER 1024
#define D_STATE 16
#define D_CONV  4
#define DT_RANK 32
#define IN_DIM  768
#define BB      8
#define LL      2048
#define MROWS   (BB * LL)   // 16384

typedef float  v2f __attribute__((ext_vector_type(2)));
typedef float  v8f __attribute__((ext_vector_type(8)));

// Low 32 bits of a generic pointer to LDS == LDS byte address (flat aperture).
__device__ __forceinline__ unsigned lds_off(const void* p) {
    return (unsigned)(unsigned long long)(uintptr_t)p;
}

// Async DMA: 16 bytes global -> LDS per lane, tracked by ASYNCcnt.
__device__ __forceinline__ void async_g2l_b128(unsigned lds_addr, const void* gptr) {
    asm volatile("global_load_async_to_lds_b128 %0, %1, off"
                 :: "v"(lds_addr), "v"((unsigned long long)(uintptr_t)gptr)
                 : "memory");
}

#define WAIT_ASYNC(n) asm volatile("s_wait_asynccnt " #n ::: "memory")

// ---------------------------------------------------------------------------
// Generic fp32 WMMA GEMM:  C[M,N] = A[M,K] (row-major, row stride lda)
//                                   @ W[N,K]^T  (+ bias[N]) (+ addend[M,N])
// grid = (N/64, M/16), block = 128 (4 waves). K % 32 == 0, N % 64 == 0.
// ---------------------------------------------------------------------------
__global__ __launch_bounds__(128) void mamba_wmma_gemm_f32(
    const float* __restrict__ A, int lda,
    const float* __restrict__ W, int K,
    const float* __restrict__ bias,
    const float* __restrict__ addend,
    float* __restrict__ C, int N)
{
    __shared__ float lsA[2][16 * 32];   // double-buffered A tile
    __shared__ float lsW[2][64 * 32];   // double-buffered W tile

    const int tid  = threadIdx.x;
    const int lane = tid & 31;
    const int wave = tid >> 5;
    const int m0   = blockIdx.y * 16;
    const int n0   = blockIdx.x * 64;
    const int hi   = lane >> 4;      // 0: lanes 0-15, 1: lanes 16-31
    const int l15  = lane & 15;

    // Per-thread staging coordinates (A: 1 x float4, W: 4 x float4)
    const int arow = tid >> 3, akq = (tid & 7) << 2;

    // Issue the 5 async b128 copies for K-tile starting at k0 into buffer buf.
    auto stage = [&](int buf, int k0) {
        async_g2l_b128(lds_off(&lsA[buf][arow * 32 + akq]),
                       &A[(size_t)(m0 + arow) * lda + k0 + akq]);
        #pragma unroll
        for (int i = 0; i < 4; ++i) {
            int idx = tid + i * 128;
            int row = idx >> 3, kq = (idx & 7) << 2;
            async_g2l_b128(lds_off(&lsW[buf][row * 32 + kq]),
                           &W[(size_t)(n0 + row) * K + k0 + kq]);
        }
    };

    v8f acc = {};
    const int nt = K >> 5;           // number of 32-wide K tiles
    stage(0, 0);

    for (int kt = 0; kt < nt; ++kt) {
        if (kt + 1 < nt) {
            stage((kt + 1) & 1, (kt + 1) << 5);   // prefetch next tile
            WAIT_ASYNC(5);                        // oldest 5 (tile kt) landed
        } else {
            WAIT_ASYNC(0);
        }
        __syncthreads();

        const float* tA = lsA[kt & 1];
        const float* tW = lsW[kt & 1];
        const int wn = wave * 16 + l15;           // W row inside 64-row tile
        #pragma unroll
        for (int kk = 0; kk < 32; kk += 4) {
            v2f a, b;
            // A 16x4: lanes 0-15 -> K {0,1}; lanes 16-31 -> K {2,3}
            a.x = tA[l15 * 32 + kk + 2 * hi + 0];
            a.y = tA[l15 * 32 + kk + 2 * hi + 1];
            // B 4x16 (= W^T tile), mirrored layout across lane halves
            b.x = tW[wn * 32 + kk + 2 * hi + 0];
            b.y = tW[wn * 32 + kk + 2 * hi + 1];
            acc = __builtin_amdgcn_wmma_f32_16x16x4_f32(
                false, a, false, b, (short)0, acc, false, false);
        }
        __syncthreads();   // all waves done with buf before it is re-filled
    }

    // C/D layout: VGPR r -> M = r (+8 for upper lane half), N = lane&15
    const int n = n0 + wave * 16 + l15;
    #pragma unroll
    for (int r = 0; r < 8; ++r) {
        int m   = m0 + r + hi * 8;
        float v = acc[r];
        if (bias)   v += bias[n];
        if (addend) v += addend[(size_t)m * N + n];
        C[(size_t)m * N + n] = v;
    }
}

// ---------------------------------------------------------------------------
// LayerNorm over rows of 512. One block (256 threads) per row.
// ---------------------------------------------------------------------------
__global__ __launch_bounds__(256) void mamba_layernorm(
    const float* __restrict__ h, const float* __restrict__ gamma,
    const float* __restrict__ beta, float* __restrict__ out)
{
    const int row = blockIdx.x;
    const int tid = threadIdx.x;
    const float* r = h + (size_t)row * D_MODEL;
    float v0 = r[tid], v1 = r[tid + 256];

    __shared__ float ssum[256], ssq[256];
    ssum[tid] = v0 + v1;
    ssq[tid]  = v0 * v0 + v1 * v1;
    __syncthreads();
    for (int s = 128; s > 0; s >>= 1) {
        if (tid < s) { ssum[tid] += ssum[tid + s]; ssq[tid] += ssq[tid + s]; }
        __syncthreads();
    }
    float mu  = ssum[0] * (1.f / D_MODEL);
    float var = ssq[0] * (1.f / D_MODEL) - mu * mu;
    float rs  = rsqrtf(var + 1e-5f);

    out[(size_t)row * D_MODEL + tid]       = (v0 - mu) * rs * gamma[tid] + beta[tid];
    out[(size_t)row * D_MODEL + tid + 256] = (v1 - mu) * rs * gamma[tid + 256] + beta[tid + 256];
}

// ---------------------------------------------------------------------------
// Depthwise causal conv1d (width 4) + bias + SiLU.
// Input: xm = xz[..., 0:1024] (row stride 2048). Output xc [B,L,1024].
// ---------------------------------------------------------------------------
__global__ __launch_bounds__(256) void mamba_conv_silu(
    const float* __restrict__ xz, const float* __restrict__ cw,
    const float* __restrict__ cb, float* __restrict__ xc, int total)
{
    for (int idx = blockIdx.x * blockDim.x + threadIdx.x; idx < total;
         idx += gridDim.x * blockDim.x) {
        int d = idx & (D_INNER - 1);
        int l = (idx >> 10) & (LL - 1);
        int b = idx >> 21;
        float acc = cb[d];
        #pragma unroll
        for (int j = 0; j < D_CONV; ++j) {
            int ll = l - (D_CONV - 1) + j;
            if (ll >= 0)
                acc += xz[((size_t)(b * LL + ll)) * (2 * D_INNER) + d] * cw[d * D_CONV + j];
        }
        xc[idx] = acc / (1.f + __expf(-acc));   // SiLU
    }
}

// ---------------------------------------------------------------------------
// Selective scan. One thread per channel d (8192 channels = 32 blocks x 256).
// State h[16] in registers; B/C staged into double-buffered 64-step LDS
// chunks via async global->LDS so chunk c+1 streams in during chunk c.
// Fuses softplus(dt), D*x skip and SiLU(z) gating.
// ---------------------------------------------------------------------------
__global__ __launch_bounds__(256) void mamba_scan(
    const float* __restrict__ dtr,   // [M,1024] pre-softplus dt
    const float* __restrict__ xc,    // [M,1024] conv output
    const float* __restrict__ xz,    // [M,2048] (z in second half)
    const float* __restrict__ xdbl,  // [M,64]   (dt_low|B|C)
    const float* __restrict__ A_log, // [1024,16]
    const float* __restrict__ Dp,    // [1024]
    float* __restrict__ y)           // [M,1024]
{
    const int b = blockIdx.x >> 2;
    const int d = ((blockIdx.x & 3) << 8) + threadIdx.x;

    float A[D_STATE];
    #pragma unroll
    for (int n = 0; n < D_STATE; ++n) A[n] = -__expf(A_log[d * D_STATE + n]);
    const float Dd = Dp[d];

    float h[D_STATE] = {};
    __shared__ float bc[2][64 * 32];         // 2 x (64 steps x (16 B + 16 C))
    const size_t rowbase = (size_t)b * LL;

    // 2048 floats per chunk = 256 threads x 2 float4 async copies
    auto stageS = [&](int buf, int c) {
        #pragma unroll
        for (int q = 0; q < 2; ++q) {
            int i  = threadIdx.x * 8 + q * 4;
            int tt = i >> 5, col = i & 31;
            async_g2l_b128(lds_off(&bc[buf][tt * 32 + col]),
                           &xdbl[(rowbase + c * 64 + tt) * 64 + DT_RANK + col]);
        }
    };

    const int nchunks = LL / 64;   // 32
    stageS(0, 0);

    for (int c = 0; c < nchunks; ++c) {
        if (c + 1 < nchunks) {
            stageS((c + 1) & 1, c + 1);   // prefetch next chunk
            WAIT_ASYNC(2);                // oldest 2 (chunk c) landed
        } else {
            WAIT_ASYNC(0);
        }
        __syncthreads();

        const float* tb = bc[c & 1];
        for (int tt = 0; tt < 64; ++tt) {
            const size_t t = rowbase + c * 64 + tt;
            float dtv = dtr[t * D_INNER + d];
            dtv = dtv > 20.f ? dtv : log1pf(__expf(dtv));   // softplus
            const float xv = xc[t * D_INNER + d];
            const float zv = xz[t * (2 * D_INNER) + D_INNER + d];
            const float dbx = dtv * xv;
            float yv = 0.f;
            #pragma unroll
            for (int n = 0; n < D_STATE; ++n) {
                float dA = __expf(dtv * A[n]);
                h[n] = dA * h[n] + dbx * tb[tt * 32 + n];
                yv  += h[n] * tb[tt * 32 + 16 + n];
            }
            yv += Dd * xv;
            y[t * D_INNER + d] = yv * (zv / (1.f + __expf(-zv)));
        }
        __syncthreads();   // done reading buf before it is re-filled
    }
}

// ---------------------------------------------------------------------------
extern "C" void kernel_launch(void* const* d_in, const int* in_sizes, int n_in,
                              void* d_out, int out_size, void* d_ws, size_t ws_size,
                              hipStream_t stream) {
    const float* x      = (const float*)d_in[0];
    const float* W_proj = (const float*)d_in[1];
    const float* b_proj = (const float*)d_in[2];
    const float* gamma  = (const float*)d_in[3];
    const float* beta   = (const float*)d_in[4];
    const float* W_in   = (const float*)d_in[5];
    const float* conv_w = (const float*)d_in[6];
    const float* conv_b = (const float*)d_in[7];
    const float* W_x    = (const float*)d_in[8];
    const float* W_dt   = (const float*)d_in[9];
    const float* b_dt   = (const float*)d_in[10];
    const float* A_log  = (const float*)d_in[11];
    const float* Dvec   = (const float*)d_in[12];
    const float* W_out  = (const float*)d_in[13];
    float* out = (float*)d_out;

    // Workspace layout (fp32)
    float* ws   = (float*)d_ws;
    float* h    = ws;                                // [M,512] residual
    float* hn   = h    + (size_t)MROWS * D_MODEL;    // [M,512]
    float* xzb  = hn   + (size_t)MROWS * D_MODEL;    // [M,2048]
    float* xcb  = xzb  + (size_t)MROWS * 2 * D_INNER;// [M,1024]
    float* xdbl = xcb  + (size_t)MROWS * D_INNER;    // [M,64]
    float* dtr  = xdbl + (size_t)MROWS * 64;         // [M,1024]
    float* yb   = dtr  + (size_t)MROWS * D_INNER;    // [M,1024]

    auto gemm = [&](const float* Ap, int lda, const float* Wp, int K,
                    const float* bp, const float* ad, float* Cp, int N) {
        dim3 grid(N / 64, MROWS / 16);
        hipLaunchKernelGGL(mamba_wmma_gemm_f32, grid, dim3(128), 0, stream,
                           Ap, lda, Wp, K, bp, ad, Cp, N);
    };

    // 1. input projection (+bias) -> h (residual)
    gemm(x, IN_DIM, W_proj, IN_DIM, b_proj, nullptr, h, D_MODEL);
    // 2. LayerNorm -> hn
    hipLaunchKernelGGL(mamba_layernorm, dim3(MROWS), dim3(256), 0, stream,
                       h, gamma, beta, hn);
    // 3. in_proj -> xz
    gemm(hn, D_MODEL, W_in, D_MODEL, nullptr, nullptr, xzb, 2 * D_INNER);
    // 4. depthwise causal conv + SiLU -> xc
    {
        int total = MROWS * D_INNER;
        hipLaunchKernelGGL(mamba_conv_silu, dim3(total / 256), dim3(256), 0, stream,
                           xzb, conv_w, conv_b, xcb, total);
    }
    // 5. x_proj -> x_dbl (dt_low|B|C)
    gemm(xcb, D_INNER, W_x, D_INNER, nullptr, nullptr, xdbl, DT_RANK + 2 * D_STATE);
    // 6. dt projection (strided A: first 32 of 64 cols) + b_dt -> dtr
    gemm(xdbl, 64, W_dt, DT_RANK, b_dt, nullptr, dtr, D_INNER);
    // 7. selective scan (softplus + skip + gate fused) -> y
    hipLaunchKernelGGL(mamba_scan, dim3((BB * D_INNER) / 256), dim3(256), 0, stream,
                       dtr, xcb, xzb, xdbl, A_log, Dvec, yb);
    // 8. out_proj with fused residual -> d_out
    gemm(yb, D_INNER, W_out, D_INNER, nullptr, h, out, D_MODEL);

    (void)in_sizes; (void)n_in; (void)out_size; (void)ws_size;
}